// MeshNodeUpdate_21998822490256
// MI455X (gfx1250) — compile-verified
//
#include <hip/hip_runtime.h>
#include <cmath>

// ---- problem constants (from reference) ----
#define MNUM      40962
#define MEMB      256
#define EEMB      256
#define NUM_EDGES 327680
#define K1        512     // MEMB + EEMB
#define N1        512
#define N2        256
#define N3        256
#define LN_EPS    1e-5f

#define ROWS_PER_BLOCK 128           // 4 waves * 32 rows
#define MBLK ((MNUM + ROWS_PER_BLOCK - 1) / ROWS_PER_BLOCK)   // 321

typedef __attribute__((ext_vector_type(2))) float v2f;
typedef __attribute__((ext_vector_type(8))) float v8f;

__device__ __forceinline__ float silu_f(float x) { return x / (1.0f + expf(-x)); }

// One K-step (K += 4) for a 32x64 wave tile:
//   accL[t] (rows 0..15) and accH[t] (rows 16..31) over 4 N-tiles of 16.
// B operand is loaded once per N-tile and feeds both row halves.
__device__ __forceinline__ void wmma8_step(v8f accL[4], v8f accH[4], v2f a0, v2f a1,
                                           const float* __restrict__ Wbase,
                                           int col0, int l16, int kdim, int k0) {
#pragma unroll
  for (int t = 0; t < 4; ++t) {
    const float* wr = Wbase + (size_t)(col0 + t * 16 + l16) * kdim + k0;
    v2f b;
    b.x = wr[0];
    b.y = wr[1];
    accL[t] = __builtin_amdgcn_wmma_f32_16x16x4_f32(
        false, a0, false, b, (short)0, accL[t], false, false);
    accH[t] = __builtin_amdgcn_wmma_f32_16x16x4_f32(
        false, a1, false, b, (short)0, accH[t], false, false);
  }
}

// Store one wave's 32x64 tile with bias (+optional SiLU), guarded to r < MNUM.
__device__ __forceinline__ void store_tile32(v8f accL[4], v8f accH[4],
                                             float* __restrict__ H,
                                             const float* __restrict__ bias,
                                             int row0, int col0, int half, int l16,
                                             int ncols, int use_silu) {
#pragma unroll
  for (int v = 0; v < 8; ++v) {
    int rL = row0 + v + 8 * half;
    int rH = rL + 16;
    if (rL < MNUM) {
#pragma unroll
      for (int t = 0; t < 4; ++t) {
        int n = col0 + t * 16 + l16;
        float val = accL[t][v] + bias[n];
        if (use_silu) val = silu_f(val);
        H[(size_t)rL * ncols + n] = val;
      }
    }
    if (rH < MNUM) {
#pragma unroll
      for (int t = 0; t < 4; ++t) {
        int n = col0 + t * 16 + l16;
        float val = accH[t][v] + bias[n];
        if (use_silu) val = silu_f(val);
        H[(size_t)rH * ncols + n] = val;
      }
    }
  }
}

// ---- zero edge accumulator + reduction scalars ----
__global__ void zero_ws_kernel(float* __restrict__ eagg, float* __restrict__ sums,
                               long long n) {
  long long i = (long long)blockIdx.x * blockDim.x + threadIdx.x;
  if (i < n) eagg[i] = 0.0f;
  if (i < 2) sums[i] = 0.0f;
}

// ---- fused: copy me_x to output + scatter-add into edge_agg (L2-resident) ----
// 64 threads/block, float4 per lane (b128 load/store + 4 f32 atomics).
__global__ void scatter_copy_kernel(const float* __restrict__ me_x,
                                    const int* __restrict__ me_col,
                                    float* __restrict__ out_mex,
                                    float* __restrict__ eagg) {
  int e = blockIdx.x;
  int f = threadIdx.x * 4;
  size_t idx = (size_t)e * EEMB + f;
  float4 v = *(const float4*)(me_x + idx);
  *(float4*)(out_mex + idx) = v;
  int col = me_col[e];
  float* dst = eagg + (size_t)col * EEMB + f;
  atomicAdd(dst + 0, v.x);
  atomicAdd(dst + 1, v.y);
  atomicAdd(dst + 2, v.z);
  atomicAdd(dst + 3, v.w);
}

// ---- GEMM1: H1 = silu([mx | edge_agg] @ W1^T + b1), A concat read from two sources ----
// A rows clamped (not masked): out-of-range rows produce garbage that is never stored,
// so the hot loop stays branch-free with EXEC all-ones for WMMA.
__global__ __launch_bounds__(128)
void gemm1_wmma_kernel(const float* __restrict__ mx, const float* __restrict__ eagg,
                       const float* __restrict__ W1, const float* __restrict__ b1,
                       float* __restrict__ H1) {
  const int lane = threadIdx.x & 31;
  const int wave = threadIdx.x >> 5;
  const int half = lane >> 4;
  const int l16  = lane & 15;
  const int row0 = blockIdx.x * ROWS_PER_BLOCK + wave * 32;
  const int col0 = blockIdx.y * 64;
  int arL = row0 + l16;       if (arL >= MNUM) arL = MNUM - 1;
  int arH = row0 + 16 + l16;  if (arH >= MNUM) arH = MNUM - 1;

  v8f accL[4] = {v8f{}, v8f{}, v8f{}, v8f{}};
  v8f accH[4] = {v8f{}, v8f{}, v8f{}, v8f{}};

  // K = 0..255 from mx
  const float* AmxL = mx + (size_t)arL * MEMB;
  const float* AmxH = mx + (size_t)arH * MEMB;
  for (int kb = 0; kb < MEMB; kb += 4) {
    int k0 = kb + 2 * half;
    v2f a0, a1;
    a0.x = AmxL[k0]; a0.y = AmxL[k0 + 1];
    a1.x = AmxH[k0]; a1.y = AmxH[k0 + 1];
    wmma8_step(accL, accH, a0, a1, W1, col0, l16, K1, k0);
  }
  // K = 256..511 from edge_agg
  const float* AeaL = eagg + (size_t)arL * EEMB;
  const float* AeaH = eagg + (size_t)arH * EEMB;
  for (int kb = 0; kb < EEMB; kb += 4) {
    int k0 = kb + 2 * half;
    v2f a0, a1;
    a0.x = AeaL[k0]; a0.y = AeaL[k0 + 1];
    a1.x = AeaH[k0]; a1.y = AeaH[k0 + 1];
    wmma8_step(accL, accH, a0, a1, W1 + MEMB, col0, l16, K1, k0);
  }
  store_tile32(accL, accH, H1, b1, row0, col0, half, l16, N1, /*silu=*/1);
}

// ---- generic GEMM: H = act(A @ W^T + b), A rows clamped to MNUM-1 ----
template <int KDIM, int NCOLS, int SILU>
__global__ __launch_bounds__(128)
void gemm_wmma_kernel(const float* __restrict__ A, const float* __restrict__ W,
                      const float* __restrict__ bias, float* __restrict__ H) {
  const int lane = threadIdx.x & 31;
  const int wave = threadIdx.x >> 5;
  const int half = lane >> 4;
  const int l16  = lane & 15;
  const int row0 = blockIdx.x * ROWS_PER_BLOCK + wave * 32;
  const int col0 = blockIdx.y * 64;
  int arL = row0 + l16;       if (arL >= MNUM) arL = MNUM - 1;
  int arH = row0 + 16 + l16;  if (arH >= MNUM) arH = MNUM - 1;

  v8f accL[4] = {v8f{}, v8f{}, v8f{}, v8f{}};
  v8f accH[4] = {v8f{}, v8f{}, v8f{}, v8f{}};

  const float* ArL = A + (size_t)arL * KDIM;
  const float* ArH = A + (size_t)arH * KDIM;
  for (int kb = 0; kb < KDIM; kb += 4) {
    int k0 = kb + 2 * half;
    v2f a0, a1;
    a0.x = ArL[k0]; a0.y = ArL[k0 + 1];
    a1.x = ArH[k0]; a1.y = ArH[k0 + 1];
    wmma8_step(accL, accH, a0, a1, W, col0, l16, KDIM, k0);
  }
  store_tile32(accL, accH, H, bias, row0, col0, half, l16, NCOLS, SILU);
}

// ---- global-LN pass 1: block partial sums of h and h^2 (4 elems/thread, b128) ----
__global__ void ln_reduce_kernel(const float* __restrict__ h3, float* __restrict__ sums) {
  __shared__ float s1[256];
  __shared__ float s2[256];
  const size_t nquads = (size_t)MNUM * MEMB / 4;
  size_t q = (size_t)blockIdx.x * 256 + threadIdx.x;
  float a = 0.0f, b = 0.0f;
  if (q < nquads) {
    float4 v = *(const float4*)(h3 + q * 4);
    a = v.x + v.y + v.z + v.w;
    b = v.x * v.x + v.y * v.y + v.z * v.z + v.w * v.w;
  }
  s1[threadIdx.x] = a;
  s2[threadIdx.x] = b;
  __syncthreads();
  for (int off = 128; off > 0; off >>= 1) {
    if (threadIdx.x < off) {
      s1[threadIdx.x] += s1[threadIdx.x + off];
      s2[threadIdx.x] += s2[threadIdx.x + off];
    }
    __syncthreads();
  }
  if (threadIdx.x == 0) {
    atomicAdd(&sums[0], s1[0]);
    atomicAdd(&sums[1], s2[0]);
  }
}

// ---- global-LN pass 2 + residual, in place over the mx_new slot of d_out ----
__global__ void ln_final_kernel(const float* __restrict__ sums,
                                const float* __restrict__ mx,
                                const float* __restrict__ gamma,
                                const float* __restrict__ beta,
                                float* __restrict__ out_mx) {
  const size_t nquads = (size_t)MNUM * MEMB / 4;
  size_t q = (size_t)blockIdx.x * 256 + threadIdx.x;
  if (q >= nquads) return;
  const float n = (float)((size_t)MNUM * MEMB);
  float mu  = sums[0] / n;
  float var = sums[1] / n - mu * mu;
  float rs  = rsqrtf(var + LN_EPS);
  size_t i = q * 4;
  float4 h = *(const float4*)(out_mx + i);
  float4 g = *(const float4*)(gamma + i);
  float4 bb = *(const float4*)(beta + i);
  float4 m = *(const float4*)(mx + i);
  float4 o;
  o.x = (h.x - mu) * rs * g.x + bb.x + m.x;
  o.y = (h.y - mu) * rs * g.y + bb.y + m.y;
  o.z = (h.z - mu) * rs * g.z + bb.z + m.z;
  o.w = (h.w - mu) * rs * g.w + bb.w + m.w;
  *(float4*)(out_mx + i) = o;
}

extern "C" void kernel_launch(void* const* d_in, const int* in_sizes, int n_in,
                              void* d_out, int out_size, void* d_ws, size_t ws_size,
                              hipStream_t stream) {
  const float* mx     = (const float*)d_in[1];
  const int*   me_i   = (const int*)d_in[2];
  const float* me_x   = (const float*)d_in[3];
  const float* W1     = (const float*)d_in[8];
  const float* b1     = (const float*)d_in[9];
  const float* W2     = (const float*)d_in[10];
  const float* b2     = (const float*)d_in[11];
  const float* W3     = (const float*)d_in[12];
  const float* b3     = (const float*)d_in[13];
  const float* gamma  = (const float*)d_in[14];
  const float* beta   = (const float*)d_in[15];

  // d_out = outputs concatenated flat in return order (indices 0..7 of inputs)
  size_t off[8];
  size_t acc = 0;
  for (int i = 0; i < 8; ++i) { off[i] = acc; acc += (size_t)in_sizes[i]; }
  float* out = (float*)d_out;
  float* out_mx  = out + off[1];
  float* out_mex = out + off[3];

  // workspace: edge_agg (MNUM x 256, later reused as H2), H1 (MNUM x 512), 2 scalars
  char* ws = (char*)d_ws;
  float* eagg = (float*)ws;
  float* H1   = (float*)(ws + (size_t)MNUM * 256 * 4);
  float* sums = (float*)(ws + (size_t)MNUM * 256 * 4 + (size_t)MNUM * 512 * 4);

  // pass-through copies (me_x handled fused with the scatter kernel)
  hipMemcpyAsync(out + off[0], d_in[0], (size_t)in_sizes[0] * 4, hipMemcpyDeviceToDevice, stream);
  hipMemcpyAsync(out + off[2], d_in[2], (size_t)in_sizes[2] * 4, hipMemcpyDeviceToDevice, stream);
  hipMemcpyAsync(out + off[4], d_in[4], (size_t)in_sizes[4] * 4, hipMemcpyDeviceToDevice, stream);
  hipMemcpyAsync(out + off[5], d_in[5], (size_t)in_sizes[5] * 4, hipMemcpyDeviceToDevice, stream);
  hipMemcpyAsync(out + off[6], d_in[6], (size_t)in_sizes[6] * 4, hipMemcpyDeviceToDevice, stream);
  hipMemcpyAsync(out + off[7], d_in[7], (size_t)in_sizes[7] * 4, hipMemcpyDeviceToDevice, stream);

  // 1) zero accumulator + LN scalars
  long long nz = (long long)MNUM * 256;
  zero_ws_kernel<<<(unsigned)((nz + 255) / 256), 256, 0, stream>>>(eagg, sums, nz);

  // 2) fused me_x copy + segment-sum scatter (float4 lanes)
  scatter_copy_kernel<<<NUM_EDGES, 64, 0, stream>>>(me_x, me_i + NUM_EDGES, out_mex, eagg);

  // 3) MLP layer 1: [mx | eagg] @ W1^T -> H1 (silu)
  dim3 g1(MBLK, N1 / 64);
  gemm1_wmma_kernel<<<g1, 128, 0, stream>>>(mx, eagg, W1, b1, H1);

  // 4) MLP layer 2: H1 @ W2^T -> H2 (silu), H2 reuses the edge_agg buffer
  dim3 g2(MBLK, N2 / 64);
  gemm_wmma_kernel<512, 256, 1><<<g2, 128, 0, stream>>>(H1, W2, b2, eagg);

  // 5) MLP layer 3: H2 @ W3^T -> h3, stored directly into the mx_new output slot
  dim3 g3(MBLK, N3 / 64);
  gemm_wmma_kernel<256, 256, 0><<<g3, 128, 0, stream>>>(eagg, W3, b3, out_mx);

  // 6) global LayerNorm stats over h3 (4 elements per thread)
  unsigned nq = (unsigned)(((size_t)MNUM * MEMB / 4 + 255) / 256);
  ln_reduce_kernel<<<nq, 256, 0, stream>>>(out_mx, sums);

  // 7) normalize + affine + residual, in place
  ln_final_kernel<<<nq, 256, 0, stream>>>(sums, mx, gamma, beta, out_mx);
}